// MFbpr_28089086116333
// MI455X (gfx1250) — compile-verified
//
#include <hip/hip_runtime.h>
#include <math.h>

typedef __attribute__((ext_vector_type(2))) float v2f;
typedef __attribute__((ext_vector_type(8))) float v8f;

#define FACTORS 64
#define REG_C 0.1f
#define INV_LN2 1.4426950408889634f

__device__ __forceinline__ float log_sigmoid_f(float x) {
    // log(sigmoid(x)) = min(x,0) - log1p(exp(-|x|))  (numerically stable)
    return fminf(x, 0.0f) - log1pf(__expf(-fabsf(x)));
}

// One wave32 per 16-row batch tile.
// A-frag (Uu, 16x4 f32) and B-frag (Vi/Vj^T, 4x16 f32) share the same lane
// addressing: lane L in [0,16) holds K-pair {4k,4k+1} of row L, lane L+16
// holds pair {4k+2,4k+3}. Chained V_WMMA_F32_16X16X4_F32 over k=0..15 gives
// the 16x16 Gram tile; its diagonal is the per-row dot product.
__global__ void bpr_wmma_kernel(const float* __restrict__ U,
                                const float* __restrict__ V,
                                const int* __restrict__ uIdx,
                                const int* __restrict__ iIdx,
                                const int* __restrict__ jIdx,
                                float* __restrict__ y_ui,
                                float* __restrict__ y_uj,
                                float* __restrict__ tilePartial,
                                int nTiles)
{
    const int lane = threadIdx.x & 31;
    const int wave = threadIdx.x >> 5;
    const int tile = blockIdx.x * (blockDim.x >> 5) + wave;
    if (tile >= nTiles) return;   // uniform per wave -> EXEC stays all-ones

    const int row  = (tile << 4) + (lane & 15);
    const int koff = (lane >> 4) << 1;   // 0 for lanes 0-15, 2 for lanes 16-31

    const float* __restrict__ Urow  = U + (size_t)uIdx[row] * FACTORS;
    const float* __restrict__ Virow = V + (size_t)iIdx[row] * FACTORS;
    const float* __restrict__ Vjrow = V + (size_t)jIdx[row] * FACTORS;

    v8f cui = {};
    v8f cuj = {};
    float ssq = 0.0f;

#pragma unroll
    for (int kk = 0; kk < FACTORS / 4; ++kk) {
        const int k = 4 * kk + koff;
        v2f a  = *(const v2f*)(Urow  + k);
        v2f bi = *(const v2f*)(Virow + k);
        v2f bj = *(const v2f*)(Vjrow + k);
        // complementary lane pairs cover each row exactly once -> exact ssq
        ssq += a.x * a.x + a.y * a.y
             + bi.x * bi.x + bi.y * bi.y
             + bj.x * bj.x + bj.y * bj.y;
        // 8 args: (neg_a, A, neg_b, B, c_mod, C, reuse_a, reuse_b)
        cui = __builtin_amdgcn_wmma_f32_16x16x4_f32(
            false, a, false, bi, (short)0, cui, false, false);
        cuj = __builtin_amdgcn_wmma_f32_16x16x4_f32(
            false, a, false, bj, (short)0, cuj, false, false);
    }

    // Diagonal of 16x16 f32 C/D: m in 0..7 -> lane m, VGPR m;
    //                            m in 8..15 -> lane m+16 (24..31), VGPR m-8.
    const bool hasDiag = (lane < 8) || (lane >= 24);
    const int  m = (lane < 8) ? lane : (lane - 16);
    const int  r = m & 7;

    float dui = 0.0f, duj = 0.0f;
#pragma unroll
    for (int q = 0; q < 8; ++q) {
        dui = (r == q) ? cui[q] : dui;
        duj = (r == q) ? cuj[q] : duj;
    }

    float ls = 0.0f;
    if (hasDiag) {
        const int b = (tile << 4) + m;
        y_ui[b] = dui;
        y_uj[b] = duj;
        ls = log_sigmoid_f(dui - duj);
    }

    // Per-lane partial of (REG*ssq - ls/ln2); reduce across the wave32.
    float p = REG_C * ssq - INV_LN2 * ls;
#pragma unroll
    for (int off = 16; off > 0; off >>= 1)
        p += __shfl_xor(p, off, 32);

    if (lane == 0)
        tilePartial[tile] = p;
}

// Deterministic fixed-order reduction of per-tile partials into the scalar
// loss slot (overwrites; never accumulates into d_out).
__global__ void bpr_reduce_kernel(const float* __restrict__ tilePartial,
                                  float* __restrict__ lossOut,
                                  int nTiles)
{
    __shared__ float smem[256];
    float acc = 0.0f;
    for (int t = threadIdx.x; t < nTiles; t += 256)
        acc += tilePartial[t];
    smem[threadIdx.x] = acc;
    __syncthreads();
#pragma unroll
    for (int s = 128; s > 0; s >>= 1) {
        if ((int)threadIdx.x < s)
            smem[threadIdx.x] += smem[threadIdx.x + s];
        __syncthreads();
    }
    if (threadIdx.x == 0)
        *lossOut = smem[0];
}

extern "C" void kernel_launch(void* const* d_in, const int* in_sizes, int n_in,
                              void* d_out, int out_size, void* d_ws, size_t ws_size,
                              hipStream_t stream) {
    const float* U  = (const float*)d_in[0];
    const float* V  = (const float*)d_in[1];
    const int*   u  = (const int*)d_in[2];
    const int*   ii = (const int*)d_in[3];
    const int*   jj = (const int*)d_in[4];

    const int B      = in_sizes[2];      // 16384
    const int nTiles = B / 16;           // 1024 (B divisible by 16)

    float* y_ui = (float*)d_out;
    float* y_uj = y_ui + B;
    float* loss = y_ui + 2 * B;
    float* partial = (float*)d_ws;       // nTiles floats of scratch

    const int wavesPerBlock = 8;         // 256 threads = 8 wave32 per block
    dim3 block(32 * wavesPerBlock);
    dim3 grid((nTiles + wavesPerBlock - 1) / wavesPerBlock);

    bpr_wmma_kernel<<<grid, block, 0, stream>>>(U, V, u, ii, jj,
                                                y_ui, y_uj, partial, nTiles);
    bpr_reduce_kernel<<<1, 256, 0, stream>>>(partial, loss, nTiles);
}